// Policy_575525618012
// MI455X (gfx1250) — compile-verified
//
#include <hip/hip_runtime.h>

#define NDIM 8192

typedef __attribute__((ext_vector_type(2))) float v2f;
typedef __attribute__((ext_vector_type(8))) float v8f;

// workspace layout (float offsets)
#define OFF_DEG   0
#define OFF_DINV  (1*NDIM)
#define OFF_S0    (2*NDIM)
#define OFF_S1    (3*NDIM)
#define OFF_Y0    (4*NDIM)
#define OFF_Y1    (5*NDIM)
#define WS_FLOATS (6*NDIM)

__global__ void zero_ws_kernel(float* __restrict__ ws) {
    int i = blockIdx.x * blockDim.x + threadIdx.x;
    if (i < WS_FLOATS) ws[i] = 0.0f;
}

// Pass 1: deg[c] = sum_r DSM[r,c] using V_WMMA_F32_16X16X4_F32 with A row0 = ones.
// Wave covers 64 columns (4 tiles of 16), block = 8 waves = 512 cols, grid.y chunks rows.
__global__ __launch_bounds__(256) void colsum_wmma_kernel(const float* __restrict__ dsm,
                                                          float* __restrict__ deg) {
    const int lane = threadIdx.x & 31;
    const int wid  = threadIdx.x >> 5;
    const int hi   = lane >> 4;      // 0: K rows {0,1}, 1: K rows {2,3}
    const int ln   = lane & 15;      // A: row m; B: col n
    const int colBase  = blockIdx.x * 512 + wid * 64 + ln;
    const int rowStart = blockIdx.y * 256;

    v2f a;                            // A[m][k]: row 0 = ones -> D[0][n] = sum_k B[k][n]
    a.x = (ln == 0) ? 1.0f : 0.0f;
    a.y = a.x;

    v8f acc0 = {}, acc1 = {}, acc2 = {}, acc3 = {};
    for (int r0 = rowStart; r0 < rowStart + 256; r0 += 4) {
        const float* p = dsm + (size_t)(r0 + 2 * hi) * NDIM + colBase;
        v2f b0, b1, b2, b3;
        b0.x = p[0];  b0.y = p[NDIM];          // rows r0+2hi, r0+2hi+1
        b1.x = p[16]; b1.y = p[NDIM + 16];
        b2.x = p[32]; b2.y = p[NDIM + 32];
        b3.x = p[48]; b3.y = p[NDIM + 48];
        acc0 = __builtin_amdgcn_wmma_f32_16x16x4_f32(false, a, false, b0, (short)0, acc0, false, false);
        acc1 = __builtin_amdgcn_wmma_f32_16x16x4_f32(false, a, false, b1, (short)0, acc1, false, false);
        acc2 = __builtin_amdgcn_wmma_f32_16x16x4_f32(false, a, false, b2, (short)0, acc2, false, false);
        acc3 = __builtin_amdgcn_wmma_f32_16x16x4_f32(false, a, false, b3, (short)0, acc3, false, false);
    }
    if (lane < 16) {                  // D row M=0 lives in acc[0], lanes 0..15
        atomicAdd(&deg[colBase +  0], acc0[0]);
        atomicAdd(&deg[colBase + 16], acc1[0]);
        atomicAdd(&deg[colBase + 32], acc2[0]);
        atomicAdd(&deg[colBase + 48], acc3[0]);
    }
}

// dinv[i] = deg>0 ? rsqrt(deg) : 0 ;  s[b][i] = dinv[i]*w1*x[b,i]
__global__ void prep_kernel(const float* __restrict__ deg, const float* __restrict__ x,
                            const float* __restrict__ w1p, float* __restrict__ dinv,
                            float* __restrict__ s0, float* __restrict__ s1) {
    int i = blockIdx.x * blockDim.x + threadIdx.x;
    if (i >= NDIM) return;
    float d  = deg[i];
    float dv = (d > 0.0f) ? rsqrtf(d) : 0.0f;
    dinv[i] = dv;
    float w1 = w1p[0];
    s0[i] = dv * w1 * x[i];
    s1[i] = dv * w1 * x[NDIM + i];
}

// Pass 2: y[b,c] += sum_r DSM[r,c]*s[b][r] ; A rows 0/1 carry s0/s1 for the 4 K-rows.
__global__ __launch_bounds__(256) void gemv_wmma_kernel(const float* __restrict__ dsm,
                                                        const float* __restrict__ s0,
                                                        const float* __restrict__ s1,
                                                        float* __restrict__ y0,
                                                        float* __restrict__ y1) {
    const int lane = threadIdx.x & 31;
    const int wid  = threadIdx.x >> 5;
    const int hi   = lane >> 4;
    const int ln   = lane & 15;
    const int colBase  = blockIdx.x * 512 + wid * 64 + ln;
    const int rowStart = blockIdx.y * 256;

    const float2* s0v = (const float2*)s0;
    const float2* s1v = (const float2*)s1;

    v8f acc0 = {}, acc1 = {}, acc2 = {}, acc3 = {};
    for (int r0 = rowStart; r0 < rowStart + 256; r0 += 4) {
        float2 p0 = s0v[(r0 >> 1) + hi];   // {s0[r0+2hi], s0[r0+2hi+1]}
        float2 p1 = s1v[(r0 >> 1) + hi];
        v2f a;
        a.x = (ln == 0) ? p0.x : ((ln == 1) ? p1.x : 0.0f);
        a.y = (ln == 0) ? p0.y : ((ln == 1) ? p1.y : 0.0f);

        const float* p = dsm + (size_t)(r0 + 2 * hi) * NDIM + colBase;
        v2f b0, b1, b2, b3;
        b0.x = p[0];  b0.y = p[NDIM];
        b1.x = p[16]; b1.y = p[NDIM + 16];
        b2.x = p[32]; b2.y = p[NDIM + 32];
        b3.x = p[48]; b3.y = p[NDIM + 48];
        acc0 = __builtin_amdgcn_wmma_f32_16x16x4_f32(false, a, false, b0, (short)0, acc0, false, false);
        acc1 = __builtin_amdgcn_wmma_f32_16x16x4_f32(false, a, false, b1, (short)0, acc1, false, false);
        acc2 = __builtin_amdgcn_wmma_f32_16x16x4_f32(false, a, false, b2, (short)0, acc2, false, false);
        acc3 = __builtin_amdgcn_wmma_f32_16x16x4_f32(false, a, false, b3, (short)0, acc3, false, false);
    }
    if (lane < 16) {                  // acc[0] = row M=0 (batch 0), acc[1] = row M=1 (batch 1)
        atomicAdd(&y0[colBase +  0], acc0[0]); atomicAdd(&y1[colBase +  0], acc0[1]);
        atomicAdd(&y0[colBase + 16], acc1[0]); atomicAdd(&y1[colBase + 16], acc1[1]);
        atomicAdd(&y0[colBase + 32], acc2[0]); atomicAdd(&y1[colBase + 32], acc2[1]);
        atomicAdd(&y0[colBase + 48], acc3[0]); atomicAdd(&y1[colBase + 48], acc3[1]);
    }
}

// softmax over a size-1 axis == 1.0 everywhere
__global__ void ones_kernel(float* __restrict__ out) {
    int i = blockIdx.x * blockDim.x + threadIdx.x;
    if (i < 2 * NDIM) out[i] = 1.0f;
}

__global__ __launch_bounds__(256) void finalize_kernel(
    const float* __restrict__ dinv, const float* __restrict__ y0, const float* __restrict__ y1,
    const float* b1p, const float* lw1p, const float* lb1p, const float* lw2p, const float* lb2p,
    const float* w2p, const float* b2p, const float* w3p, const float* b3p,
    const float* wvp, const float* bvp, float* __restrict__ out_sv) {
    __shared__ float sm0[256];
    __shared__ float sm1[256];
    const float b1 = b1p[0], lw1 = lw1p[0], lb1 = lb1p[0], lw2 = lw2p[0], lb2 = lb2p[0];
    float a0 = 0.0f, a1 = 0.0f;
    for (int c = threadIdx.x; c < NDIM; c += 256) {
        float dv = dinv[c];
        float h0 = fmaxf(dv * y0[c] + b1, 0.0f);
        float h1 = fmaxf(dv * y1[c] + b1, 0.0f);
        h0 = fmaxf(lw1 * h0 + lb1, 0.0f);
        h1 = fmaxf(lw1 * h1 + lb1, 0.0f);
        h0 = fmaxf(lw2 * h0 + lb2, 0.0f);
        h1 = fmaxf(lw2 * h1 + lb2, 0.0f);
        a0 += h0; a1 += h1;
    }
    sm0[threadIdx.x] = a0; sm1[threadIdx.x] = a1;
    __syncthreads();
    for (int s = 128; s > 0; s >>= 1) {
        if ((int)threadIdx.x < s) {
            sm0[threadIdx.x] += sm0[threadIdx.x + s];
            sm1[threadIdx.x] += sm1[threadIdx.x + s];
        }
        __syncthreads();
    }
    if (threadIdx.x == 0) {
        const float w2 = w2p[0], b2 = b2p[0], w3 = w3p[0], b3 = b3p[0], wv = wvp[0], bv = bvp[0];
        float m0 = sm0[0] * (1.0f / NDIM);
        float m1 = sm1[0] * (1.0f / NDIM);
        float h40 = fmaxf(w2 * m0 + b2, 0.0f);
        float h41 = fmaxf(w2 * m1 + b2, 0.0f);
        float h50 = fmaxf(w3 * h40 + b3, 0.0f);
        float h51 = fmaxf(w3 * h41 + b3, 0.0f);
        out_sv[0] = fmaxf(h50, h51) * wv + bv;
    }
}

extern "C" void kernel_launch(void* const* d_in, const int* in_sizes, int n_in,
                              void* d_out, int out_size, void* d_ws, size_t ws_size,
                              hipStream_t stream) {
    const float* x   = (const float*)d_in[0];
    const float* dsm = (const float*)d_in[1];
    const float* w1  = (const float*)d_in[2];
    const float* b1  = (const float*)d_in[3];
    const float* lw1 = (const float*)d_in[4];
    const float* lb1 = (const float*)d_in[5];
    const float* lw2 = (const float*)d_in[6];
    const float* lb2 = (const float*)d_in[7];
    const float* w2  = (const float*)d_in[8];
    const float* b2  = (const float*)d_in[9];
    const float* w3  = (const float*)d_in[10];
    const float* b3  = (const float*)d_in[11];
    // d_in[12]=wa, d_in[13]=ba: dead (softmax over singleton axis)
    const float* wvp = (const float*)d_in[14];
    const float* bvp = (const float*)d_in[15];

    float* out = (float*)d_out;
    float* ws  = (float*)d_ws;

    float* deg  = ws + OFF_DEG;
    float* dinv = ws + OFF_DINV;
    float* s0   = ws + OFF_S0;
    float* s1   = ws + OFF_S1;
    float* y0   = ws + OFF_Y0;
    float* y1   = ws + OFF_Y1;

    zero_ws_kernel<<<(WS_FLOATS + 255) / 256, 256, 0, stream>>>(ws);

    dim3 grid(NDIM / 512, 32);  // 16 col-blocks x 32 row-chunks = 512 blocks
    colsum_wmma_kernel<<<grid, 256, 0, stream>>>(dsm, deg);

    prep_kernel<<<NDIM / 256, 256, 0, stream>>>(deg, x, w1, dinv, s0, s1);

    gemv_wmma_kernel<<<grid, 256, 0, stream>>>(dsm, s0, s1, y0, y1);

    ones_kernel<<<(2 * NDIM + 255) / 256, 256, 0, stream>>>(out);

    finalize_kernel<<<1, 256, 0, stream>>>(dinv, y0, y1,
                                           b1, lw1, lb1, lw2, lb2,
                                           w2, b2, w3, b3, wvp, bvp,
                                           out + 2 * NDIM);
}